// decoder_4956392259723
// MI455X (gfx1250) — compile-verified
//
#include <hip/hip_runtime.h>
#include <hip/hip_bf16.h>

typedef __attribute__((ext_vector_type(2))) float v2f;
typedef __attribute__((ext_vector_type(8))) float v8f;

#define NUM_HEADS    8
#define NUM_CLUSTERS 200
#define S_PAD        208   // 13 tiles of 16

// float offsets inside workspace
#define WOFF_W    0        // 8 floats: softmax weights
#define WOFF_COEF 8        // S_PAD float4s: {a, b, c, 0} per cluster

// ---------------------------------------------------------------------------
// Prep: softmax(w) -> ws[0..7]; per-cluster quadratic coeffs packed as
// float4 {a,b,c,0} (zero-padded to 208 clusters) so the WMMA B-fragment is a
// single unconditional b64 load per lane: half0 lanes read {a,b} = K0,K1,
// half1 lanes read {c,0} = K2,K3.
// score = (U - d)(d - L)/(r^2+1) = a*d^2 + b*d + c with
//   inv = 1/(r^2+1), a = -inv, b = (U+L)*inv, c = -U*L*inv
// ---------------------------------------------------------------------------
__global__ __launch_bounds__(256) void prep_kernel(const float* __restrict__ w,
                                                   const float* __restrict__ r_dist,
                                                   float* __restrict__ ws) {
    __shared__ float sh[256];
    __shared__ float shw[NUM_HEADS];
    const int t = threadIdx.x;
    if (t < NUM_HEADS) shw[t] = w[t];
    const float rv = (t < NUM_CLUSTERS) ? r_dist[t] : 0.0f;
    const float ar = fabsf(rv);
    sh[t] = ar;
    __syncthreads();

    if (t < NUM_HEADS) {
        float m = shw[0];
#pragma unroll
        for (int i = 1; i < NUM_HEADS; ++i) m = fmaxf(m, shw[i]);
        float sum = 0.0f;
#pragma unroll
        for (int i = 0; i < NUM_HEADS; ++i) sum += expf(shw[i] - m);
        ws[WOFF_W + t] = expf(shw[t] - m) / sum;
    }

    // Hillis-Steele inclusive scan over 256 slots (only first 200 nonzero)
    for (int off = 1; off < 256; off <<= 1) {
        const float v = (t >= off) ? sh[t - off] : 0.0f;
        __syncthreads();
        sh[t] += v;
        __syncthreads();
    }

    if (t < S_PAD) {
        float4 cf = make_float4(0.0f, 0.0f, 0.0f, 0.0f);
        if (t < NUM_CLUSTERS) {
            const float incl  = sh[t];            // inclusive cumsum of |r|
            const float lower = incl - ar;        // exclusive cumsum
            const float upper = incl + 0.1f;
            const float inv   = 1.0f / (rv * rv + 1.0f);
            cf.x = -inv;                          // a
            cf.y = (upper + lower) * inv;         // b
            cf.z = -(upper * lower) * inv;        // c
        }
        ((float4*)(ws + WOFF_COEF))[t] = cf;
    }
}

// ---------------------------------------------------------------------------
// Main kernel: one wave32 handles 16 edges.
//   lane = part*16 + m ; m = edge-in-tile (A rows / B cols), part = lane half
//   part selects heads 0-3 vs 4-7 for the norm phase, and K{0,1} vs K{2,3}
//   for the WMMA fragments.
// A fragment (16x4 f32): rows = edges, K = (d^2, d, 1, 0)
// B fragment (4x16 f32): K rows = (a[s], b[s], c[s], 0)
// D tile: VGPR r, lanes 0-15 -> edge r; lanes 16-31 -> edge r+8.
// ---------------------------------------------------------------------------
__global__ __launch_bounds__(256) void edge_score_kernel(
    const float* __restrict__ h,
    const int*  __restrict__ src,
    const int*  __restrict__ dst,
    const float* __restrict__ ws,
    float* __restrict__ score,
    float* __restrict__ stdo,
    int n_edges) {
    const int lane   = threadIdx.x & 31;
    const int wv     = blockIdx.x * 8 + (threadIdx.x >> 5);
    const int e_base = wv << 4;
    if (e_base >= n_edges) return;           // wave-uniform exit (EXEC stays full)

    const int part = lane >> 4;              // 0 or 1
    const int m    = lane & 15;
    const int e    = e_base + m;

    const int sn = src[e];
    const int dn = dst[e];

    // h layout: [node][head][feat] f32 -> 32 float4 per node; this lane's 4 heads
    const float4* hf = (const float4*)h;
    const float4* ps = hf + (size_t)sn * 32 + part * 16;
    const float4* pd = hf + (size_t)dn * 32 + part * 16;

    float n2[4];
#pragma unroll
    for (int hh = 0; hh < 4; ++hh) {
        float acc = 0.0f;
#pragma unroll
        for (int q = 0; q < 4; ++q) {
            const float4 a = ps[hh * 4 + q];
            const float4 b = pd[hh * 4 + q];
            const float dx = b.x - a.x, dy = b.y - a.y;
            const float dz = b.z - a.z, dw = b.w - a.w;
            acc = fmaf(dx, dx, fmaf(dy, dy, fmaf(dz, dz, fmaf(dw, dw, acc))));
        }
        n2[hh] = sqrtf(acc);
    }

    // softmax weights for this lane's 4 heads
    const float4 wvec = ((const float4*)(ws + WOFF_W))[part];
    const float s1 = n2[0] * wvec.x + n2[1] * wvec.y + n2[2] * wvec.z + n2[3] * wvec.w;
    const float dist = s1 + __shfl_xor(s1, 16, 32);   // combine the two head-halves

    float s2 = 0.0f;
#pragma unroll
    for (int i = 0; i < 4; ++i) {
        const float td = n2[i] - dist;
        s2 = fmaf(td, td, s2);
    }
    const float var = s2 + __shfl_xor(s2, 16, 32);
    if (part == 0)
        __builtin_nontemporal_store(sqrtf(var * 0.125f), stdo + e);

    // A fragment: lanes 0-15 hold K0=d^2 (vgpr0), K1=d (vgpr1);
    //             lanes 16-31 hold K2=1, K3=0.
    v2f A;
    A.x = part ? 1.0f : dist * dist;
    A.y = part ? 0.0f : dist;

    // packed coeffs: lane reads float2 = half0 ? {a,b} : {c,0}  (K pair)
    const v2f* coef = (const v2f*)(ws + WOFF_COEF);
    const int coefIdx = part;                // + sc*2 per tile
    const int mrow = e_base + part * 8;

    for (int tile = 0; tile < S_PAD / 16; ++tile) {
        const int sc = tile * 16 + m;
        const v2f B = coef[sc * 2 + coefIdx];   // single b64 load, no divergence
        v8f C = {};
        v8f D = __builtin_amdgcn_wmma_f32_16x16x4_f32(
            /*neg_a=*/false, A, /*neg_b=*/false, B,
            /*c_mod=*/(short)0, C, /*reuse_a=*/false, /*reuse_b=*/false);
        if (sc < NUM_CLUSTERS) {
            float* op = score + (size_t)mrow * NUM_CLUSTERS + sc;
#pragma unroll
            for (int r = 0; r < 8; ++r)
                __builtin_nontemporal_store(D[r], op + (size_t)r * NUM_CLUSTERS);
        }
    }
}

// ---------------------------------------------------------------------------
extern "C" void kernel_launch(void* const* d_in, const int* in_sizes, int n_in,
                              void* d_out, int out_size, void* d_ws, size_t ws_size,
                              hipStream_t stream) {
    const float* h  = (const float*)d_in[0];
    const float* w  = (const float*)d_in[1];
    const float* r  = (const float*)d_in[2];
    const int* src  = (const int*)d_in[3];
    const int* dst  = (const int*)d_in[4];
    const int n_edges = in_sizes[3];

    float* ws    = (float*)d_ws;
    float* score = (float*)d_out;
    float* stdo  = score + (size_t)n_edges * NUM_CLUSTERS;

    prep_kernel<<<1, 256, 0, stream>>>(w, r, ws);

    const int waves  = (n_edges + 15) / 16;
    const int blocks = (waves + 7) / 8;   // 8 waves (256 threads) per block
    edge_score_kernel<<<blocks, 256, 0, stream>>>(h, src, dst, ws, score, stdo, n_edges);
}